// quad_norm_18202071400813
// MI455X (gfx1250) — compile-verified
//
#include <hip/hip_runtime.h>
#include <hip/hip_bf16.h>

// Dynamics constants (match reference)
#define K_DT     0.05f
#define K_ALPHA1 1.0f
#define K_ALPHA2 1.0f
#define K_BETA1  0.5f
#define K_BETA2  0.5f

// ---------------------------------------------------------------------------
// Prep: batch-uniform reduction of ATP.  ATPDh has only columns 0,2 nonzero:
//   col_k[i] = ATP[i][1]*Dh[1][k] + ATP[i][3]*Dh[3][k]
// so ||ATP@Dh||_F^2 = s11*(Dh10^2+Dh12^2) + 2*s13*(Dh10*Dh30+Dh12*Dh32)
//                   + s33*(Dh30^2+Dh32^2), with
//   s11 = sum_i ATP[i][1]^2, s13 = sum_i ATP[i][1]*ATP[i][3], s33 = sum_i ATP[i][3]^2.
// One thread computes {s11, 2*s13, s33} into d_ws once per launch (deterministic).
// ---------------------------------------------------------------------------
__global__ void quad_norm_prep_kernel(const float* __restrict__ ATP,
                                      float* __restrict__ ws)
{
    if (threadIdx.x == 0 && blockIdx.x == 0) {
        float s11 = 0.0f, s13 = 0.0f, s33 = 0.0f;
#pragma unroll
        for (int i = 0; i < 4; ++i) {
            const float t = ATP[4 * i + 1];
            const float u = ATP[4 * i + 3];
            s11 = fmaf(t, t, s11);
            s13 = fmaf(t, u, s13);
            s33 = fmaf(u, u, s33);
        }
        ws[0] = s11;
        ws[1] = 2.0f * s13;
        ws[2] = s33;
    }
}

// ---------------------------------------------------------------------------
// Main: out[i] = r - ( 2*sqrt(||ATP@Dh||_F^2) + P_norm*||Dh||_F^2 )
// Dh = (I + dt*J(x)) - A is sparse: nonzeros only at (1,0),(1,2),(3,0),(3,2).
// Memory-bound: 20 B/sample vs ~32 VALU ops/sample -> ~1.7 us floor @ 23.3 TB/s.
// ---------------------------------------------------------------------------
__global__ __launch_bounds__(256) void quad_norm_18202071400813_kernel(
    const float* __restrict__ x,
    const float* __restrict__ A,
    const float* __restrict__ Pn,
    const float* __restrict__ rr,
    const float* __restrict__ ws,
    float* __restrict__ out,
    int B)
{
    const int i = blockIdx.x * blockDim.x + threadIdx.x;

    // Uniform constant loads -> s_load_b32 (SALU side, no VGPR pressure).
    const float A10 = A[4],  A12 = A[6];
    const float A30 = A[12], A32 = A[14];
    const float P_norm = Pn[0];
    const float r      = rr[0];
    const float s11    = ws[0];
    const float s13x2  = ws[1];
    const float s33    = ws[2];

    if (i < B) {
        // Only bytes 0-3 and 8-11 of each 16B sample are needed; compiler
        // narrows this to two coalesced global_load_b32 per lane.
        const float4 xv = reinterpret_cast<const float4*>(x)[i];

        // gfx1250 prefetch path (global_prefetch_b8): stream ahead into GL2.
        // Speculative prefetch past the buffer end is silently dropped.
        __builtin_prefetch(x + 4ull * (unsigned)i + 16384, 0, 3);

        const float x1 = xv.x;
        const float x3 = xv.z;

        // Hardware transcendentals (v_cos_f32); |args| < ~9 -> ~1 ulp.
        const float cx1 = __cosf(x1);
        const float cx3 = __cosf(x3);
        const float c13 = __cosf(x1 - x3);

        // Nonzero Dh entries.  dt*beta1 == dt*beta2 == 0.025.
        const float dtbc = (K_DT * K_BETA1) * c13;
        const float Dh12 = dtbc - A12;
        const float Dh30 = dtbc - A30;
        const float Dh10 = fmaf(-(K_DT * K_ALPHA1), cx1, -dtbc) - A10;
        const float Dh32 = fmaf(-(K_DT * K_ALPHA2), cx3, -dtbc) - A32;

        // u,v,w reductions shared by both norms.
        const float u = fmaf(Dh10, Dh10, Dh12 * Dh12);
        const float v = fmaf(Dh30, Dh30, Dh32 * Dh32);
        const float w = fmaf(Dh10, Dh30, Dh12 * Dh32);

        const float dns = u + v;                                  // ||Dh||^2
        const float ans = fmaf(s11, u, fmaf(s33, v, s13x2 * w));  // ||ATP@Dh||^2

        const float q = 2.0f * __builtin_amdgcn_sqrtf(ans);
        out[i] = r - fmaf(P_norm, dns, q);
    }
}

extern "C" void kernel_launch(void* const* d_in, const int* in_sizes, int n_in,
                              void* d_out, int out_size, void* d_ws, size_t ws_size,
                              hipStream_t stream) {
    const float* x   = (const float*)d_in[0];
    const float* A   = (const float*)d_in[1];
    const float* ATP = (const float*)d_in[2];
    const float* Pn  = (const float*)d_in[3];
    const float* r   = (const float*)d_in[4];
    float* out = (float*)d_out;
    float* ws  = (float*)d_ws;

    const int B = in_sizes[0] / 4;          // x is [B,4] float32
    const int threads = 256;                // 8 wave32 per block
    const int blocks = (B + threads - 1) / threads;

    // Stage 1: uniform ATP reduction (one tiny block, same stream -> ordered).
    quad_norm_prep_kernel<<<1, 32, 0, stream>>>(ATP, ws);

    // Stage 2: streaming evaluation.
    quad_norm_18202071400813_kernel<<<blocks, threads, 0, stream>>>(
        x, A, Pn, r, ws, out, B);
}